// Model_36610301231424
// MI455X (gfx1250) — compile-verified
//
#include <hip/hip_runtime.h>
#include <math.h>

typedef __attribute__((ext_vector_type(2))) float v2f;
typedef __attribute__((ext_vector_type(8))) float v8f;

#define DD   96
#define DD2  192
#define HH   64
#define NBAT 32
#define TTOK 128
#define NROW (NBAT * TTOK)     /* 4096 */
#define W1I  (DD2 * HH)        /* 12288 floats per i */
#define HISTROW (DD * DD2)     /* 18432 floats per n row of hist_jac */
#define GSTRIDE 66             /* even padded stride: b64-aligned G reads */
#define SLOPE 0.2f

__device__ __forceinline__ float mask_scale(const float* __restrict__ alphas,
                                            int i, int d) {
    if (d < DD) return 1.0f;
    const int dp = d - DD;
    return (dp < i) ? alphas[i * DD + dp] : (dp == i ? 1.0f : 0.0f);
}

// ---------------------------------------------------------------------------
// Kernel 1: fold triangular alpha-mask into W1 AND pair-interleave for WMMA:
//   W1effP[i][(p*64+h)*2 + q] = W1[i, 2p+q, h] * s(i, 2p+q),  p = d/2, q = d&1
// so a phase-1 B fragment (rows d, d+1 at column h) is one aligned b64 in LDS.
// ---------------------------------------------------------------------------
__global__ void prep_w1eff(const float* __restrict__ W1,
                           const float* __restrict__ alphas,
                           float* __restrict__ W1effP) {
    const int bid = blockIdx.x;          // 0 .. 96*96-1  -> (i, p)
    const int i = bid / (DD2 / 2);
    const int p = bid - i * (DD2 / 2);
    const int h = threadIdx.x;           // blockDim.x == 64 == HH
    const int d0 = 2 * p;

    const float s0 = mask_scale(alphas, i, d0);
    const float s1 = mask_scale(alphas, i, d0 + 1);

    const float w0 = W1[((size_t)i * DD2 + d0) * HH + h] * s0;
    const float w1 = W1[((size_t)i * DD2 + d0 + 1) * HH + h] * s1;

    v2f* dst = (v2f*)(W1effP + (size_t)i * W1I);
    v2f out; out.x = w0; out.y = w1;
    dst[p * HH + h] = out;               // coalesced b64 stores
}

// ---------------------------------------------------------------------------
// Kernel 2: main fused kernel.  grid = (96 i-values, 64 n-chunks), block = 128.
// Each wave handles one 16-row n-tile for one i.
//  Phase 1: H = Z @ W1eff_i   (K=192, 48 wmma steps x 4 col tiles)
//  Epilogue: leaky-relu, residual row-sum, G = dact*W2 -> LDS (transpose relay)
//  Phase 2: pdd = G @ W1_i^T  (12 d-tiles x 16 wmma steps), diagonal -> logdet
// ---------------------------------------------------------------------------
__launch_bounds__(128)
__global__ void flow_main(const float* __restrict__ x,
                          const float* __restrict__ W1,
                          const float* __restrict__ b1,
                          const float* __restrict__ W2,
                          const float* __restrict__ b2,
                          const float* __restrict__ W1effP,
                          float* __restrict__ out_res,
                          float* __restrict__ out_hist,
                          float* __restrict__ partials) {
    __shared__ __align__(16) float lds[W1I];   // 48 KB: W1effP_i, reused for G
    __shared__ float lred[4];

    const int tid    = threadIdx.x;
    const int wave   = tid >> 5;
    const int lane   = tid & 31;
    const int lane15 = lane & 15;
    const int half   = lane >> 4;       // 0: lanes 0-15, 1: lanes 16-31
    const int koff   = half * 2;        // K sub-offset per fp32 wmma A/B layout
    const int i      = blockIdx.x;      // row index 0..95
    const int wg     = blockIdx.y;      // n-chunk 0..63 (64 rows each)
    const int n0     = wg * 64 + wave * 16;

    // ---- stage W1effP_i (48KB) into LDS with b128 copies ----
    {
        const float4* src = (const float4*)(W1effP + (size_t)i * W1I);
        float4* dst = (float4*)lds;
        for (int j = tid; j < W1I / 4; j += 128) dst[j] = src[j];
    }
    __syncthreads();

    // ---- per-lane A-row pointer: Z[r,:] = x_flat[(r + r/128)*96 ..] (192 contig) ----
    const int r = n0 + lane15;
    const float* arow = x + (size_t)(r + (r >> 7)) * DD;

    const v8f v8z = {0.f, 0.f, 0.f, 0.f, 0.f, 0.f, 0.f, 0.f};
    v8f acc[4] = {v8z, v8z, v8z, v8z};

    // ---- Phase 1: H(16x64) += A(16x4) * B(4x16) over K=192 ----
    for (int k = 0; k < 48; ++k) {
        const int dd = 4 * k + koff;                   // even
        const v2f a = *(const v2f*)(arow + dd);        // global b64, 8B aligned
        const float* brow = lds + (size_t)(dd >> 1) * (2 * HH);  // pair-row base
#pragma unroll
        for (int t = 0; t < 4; ++t) {
            const int col = t * 16 + lane15;
            const v2f b = *(const v2f*)(brow + 2 * col);   // single ds_load_b64
            acc[t] = __builtin_amdgcn_wmma_f32_16x16x4_f32(
                false, a, false, b, (short)0, acc[t], false, false);
        }
    }

    __syncthreads();   // all waves done reading W1effP_i before G overwrites LDS

    // ---- Epilogue: activation, residual reduction, G -> LDS ----
    float* Gl = lds + wave * (16 * GSTRIDE);   // 4224B per wave, 16.9KB total
    const float b2i = b2[i];
    float rsum[8] = {0.f, 0.f, 0.f, 0.f, 0.f, 0.f, 0.f, 0.f};
    float lsum = 0.f;

#pragma unroll
    for (int t = 0; t < 4; ++t) {
        const float w2v = W2[i * HH + t * 16 + lane15];
        const float b1v = b1[i * HH + t * 16 + lane15];
#pragma unroll
        for (int v = 0; v < 8; ++v) {
            const float hq = acc[t][v] + b1v;          // C layout: row v+8*half
            const bool pos = hq > 0.f;
            rsum[v] += (pos ? hq : SLOPE * hq) * w2v;
            Gl[(v + half * 8) * GSTRIDE + t * 16 + lane15] = (pos ? 1.f : SLOPE) * w2v;
        }
    }

    // residual: reduce each row over its 16-lane half, lane 0/16 writes
#pragma unroll
    for (int v = 0; v < 8; ++v) {
        float s = rsum[v];
        s += __shfl_xor(s, 1, 32);
        s += __shfl_xor(s, 2, 32);
        s += __shfl_xor(s, 4, 32);
        s += __shfl_xor(s, 8, 32);
        if (lane15 == 0)
            out_res[(size_t)(n0 + v + half * 8) * DD + i] = s + b2i;
    }

    // ---- load all A-frags of G (16x64) into registers (same-wave LDS, ordered) ----
    v2f ga[16];
#pragma unroll
    for (int kk = 0; kk < 16; ++kk) {
        const int hh = 4 * kk + koff;                  // even
        ga[kk] = *(const v2f*)(Gl + lane15 * GSTRIDE + hh);  // aligned ds_load_b64
    }

    // ---- Phase 2: pdd(16x192) = G(16x64) @ W1_i^T ----
    const float* W1i = W1 + (size_t)i * W1I;
    const int dspec   = DD + i;
    const int dt_spec = dspec >> 4;
    const int nspec   = dspec & 15;

    for (int dt = 0; dt < 12; ++dt) {
        v8f c = v8z;
        const int dcol = dt * 16 + lane15;
        const float* wrow = W1i + dcol * HH;           // B[K=h, N=dcol] = W1[i,dcol,h]
#pragma unroll
        for (int kk = 0; kk < 16; ++kk) {
            const v2f b = *(const v2f*)(wrow + 4 * kk + koff);   // global b64 (L2-hot)
            c = __builtin_amdgcn_wmma_f32_16x16x4_f32(
                false, ga[kk], false, b, (short)0, c, false, false);
        }
        // diagonal element (d = 96+i): take log|.|, zero it in hist_jac
        if (dt == dt_spec && lane15 == nspec) {
#pragma unroll
            for (int v = 0; v < 8; ++v) {
                lsum += logf(fabsf(c[v]));
                c[v] = 0.f;
            }
        }
        const size_t obase = (size_t)(n0 + half * 8) * HISTROW + (size_t)i * DD2 + dcol;
#pragma unroll
        for (int v = 0; v < 8; ++v)
            out_hist[obase + (size_t)v * HISTROW] = c[v];
    }

    // ---- log-det: wave reduce (non-holders carry 0), block combine, one partial ----
    lsum += __shfl_xor(lsum, 1, 32);
    lsum += __shfl_xor(lsum, 2, 32);
    lsum += __shfl_xor(lsum, 4, 32);
    lsum += __shfl_xor(lsum, 8, 32);
    lsum += __shfl_xor(lsum, 16, 32);
    if (lane == 0) lred[wave] = lsum;
    __syncthreads();
    if (tid == 0)
        partials[i * 64 + wg] = lred[0] + lred[1] + lred[2] + lred[3];
}

// ---------------------------------------------------------------------------
// Kernel 3: deterministic fixed-order reduction of log-det partials per batch.
// Each wg covers 64 rows => batch b owns wg = 2b and 2b+1.
// ---------------------------------------------------------------------------
__global__ void reduce_logdet(const float* __restrict__ partials,
                              float* __restrict__ out_ld) {
    const int b = threadIdx.x;    // 32 threads
    float s = 0.f;
    for (int i = 0; i < DD; ++i)
        s += partials[i * 64 + 2 * b] + partials[i * 64 + 2 * b + 1];
    out_ld[b] = s;
}

// ---------------------------------------------------------------------------
extern "C" void kernel_launch(void* const* d_in, const int* in_sizes, int n_in,
                              void* d_out, int out_size, void* d_ws, size_t ws_size,
                              hipStream_t stream) {
    const float* x      = (const float*)d_in[0];
    const float* alphas = (const float*)d_in[1];
    const float* W1     = (const float*)d_in[2];
    const float* b1     = (const float*)d_in[3];
    const float* W2     = (const float*)d_in[4];
    const float* b2     = (const float*)d_in[5];

    float* out      = (float*)d_out;
    float* out_res  = out;                       // 4096*96 = 393216 floats
    float* out_ld   = out + (size_t)NROW * DD;   // 32 floats
    float* out_hist = out_ld + NBAT;             // 4096*96*192 floats

    float* w1effP   = (float*)d_ws;              // 96*192*64 floats (4.7 MB)
    float* partials = w1effP + (size_t)DD * W1I; // 96*64 floats

    prep_w1eff<<<dim3(DD * (DD2 / 2)), HH, 0, stream>>>(W1, alphas, w1effP);
    flow_main<<<dim3(DD, 64), 128, 0, stream>>>(x, W1, b1, W2, b2, w1effP,
                                                out_res, out_hist, partials);
    reduce_logdet<<<1, 32, 0, stream>>>(partials, out_ld);
}